// Capsule_1477468750250
// MI455X (gfx1250) — compile-verified
//
#include <hip/hip_runtime.h>
#include <hip/hip_bf16.h>

typedef __attribute__((ext_vector_type(2))) float v2f;
typedef __attribute__((ext_vector_type(8))) float v8f;

// Problem constants
constexpr int kB = 512;           // batch
constexpr int kC = 1152;          // in_channel
constexpr int kU = 10;            // num_unit
constexpr int kO = 16;            // unit_size
constexpr int kI = 8;             // in_unit
constexpr int kK = kC * kI;       // fused contraction dim = 9216  (k = i*kC + c)
constexpr int kUO = kU * kO;      // 160
constexpr int kS1 = kK / 4;       // 2304 K-steps for GEMM1
constexpr int kS2 = kB / 4;       // 128  K-steps for GEMM2 (contraction over b)

// ---------------------------------------------------------------------------
// Generic zero fill
// ---------------------------------------------------------------------------
__global__ void zero_kernel(float* p, int n) {
    int t = blockIdx.x * blockDim.x + threadIdx.x;
    if (t < n) p[t] = 0.0f;
}

// ---------------------------------------------------------------------------
// One-time: pack x into WMMA B-fragment order for GEMM1.
//   xfrag[s][bt][lane][j] = x[b][k],  k = 4s + 2*(lane>>4) + j,
//                                     b = 16*bt + (lane&15)
// Each wave-lane in GEMM1 then loads its whole fragment with ONE b64 load,
// and 32 lanes cover 256 contiguous bytes.
// ---------------------------------------------------------------------------
__global__ void pack_xfrag_kernel(const float* __restrict__ x, float* __restrict__ xfrag) {
    int t = blockIdx.x * blockDim.x + threadIdx.x;          // kS1*32*32*2 = 4,718,592
    if (t >= kS1 * 32 * 32 * 2) return;
    int j    = t & 1;
    int lane = (t >> 1) & 31;
    int bt   = (t >> 6) & 31;
    int s    = t >> 11;
    int k = s * 4 + ((lane >> 4) << 1) + j;
    int b = bt * 16 + (lane & 15);
    xfrag[t] = x[(size_t)b * kK + k];
}

// ---------------------------------------------------------------------------
// c_coef[c][u] = softmax_u(b_log[c][u])
// ---------------------------------------------------------------------------
__global__ void softmax_kernel(const float* __restrict__ b_log, float* __restrict__ c_coef) {
    int c = blockIdx.x * blockDim.x + threadIdx.x;
    if (c >= kC) return;
    float m = -3.4e38f;
    float v[kU];
    for (int u = 0; u < kU; ++u) { v[u] = b_log[c * kU + u]; m = fmaxf(m, v[u]); }
    float s = 0.0f;
    for (int u = 0; u < kU; ++u) { v[u] = __expf(v[u] - m); s += v[u]; }
    float inv = 1.0f / s;
    for (int u = 0; u < kU; ++u) c_coef[c * kU + u] = v[u] * inv;
}

// ---------------------------------------------------------------------------
// Per-iteration: pack the coupling-scaled weights into WMMA A-fragment order.
//   afrag[u][s][lane][j] = c_coef[c][u] * W[c][u][o][i]
//       with k = 4s + 2*(lane>>4) + j,  o = lane&15,  c = k%kC,  i = k/kC
// ---------------------------------------------------------------------------
__global__ void pack_afrag_kernel(const float* __restrict__ W, const float* __restrict__ c_coef,
                                  float* __restrict__ afrag) {
    int t = blockIdx.x * blockDim.x + threadIdx.x;          // kU*kS1*32*2 = 1,474,560
    if (t >= kU * kS1 * 32 * 2) return;
    int j    = t & 1;
    int lane = (t >> 1) & 31;
    int s    = (t >> 6) % kS1;
    int u    = t / (kS1 * 64);
    int k = s * 4 + ((lane >> 4) << 1) + j;
    int o = lane & 15;
    int c = k % kC;
    int i = k / kC;
    afrag[t] = c_coef[c * kU + u] * W[((c * kU + u) * kO + o) * kI + i];
}

// ---------------------------------------------------------------------------
// GEMM1:  s_t[u][o][b] += sum_k A[u](o,k) * x(k,b)
// One wave per (u, 64-wide b-slab, K-chunk).  BN=4 register blocking:
// per K-step = 1 A-frag b64 + 4 B-frag b64 -> 4x v_wmma_f32_16x16x4_f32.
// Split-K (16 chunks of 576 K) combined with global f32 atomics.
// ---------------------------------------------------------------------------
__global__ __launch_bounds__(32) void gemm1_kernel(const float* __restrict__ afrag,
                                                   const float* __restrict__ xfrag,
                                                   float* __restrict__ s_t) {
    int u    = blockIdx.x >> 7;           // grid = 10 * 8 * 16 = 1280 waves
    int rem  = blockIdx.x & 127;
    int btg  = rem >> 4;                  // 8 slabs of 64 over B=512
    int kc   = rem & 15;                  // 16 K-chunks of 144 steps
    int lane = threadIdx.x;

    const v2f* Af = (const v2f*)afrag + (size_t)u * kS1 * 32;
    const v2f* Xf = (const v2f*)xfrag;

    v8f acc0 = {}, acc1 = {}, acc2 = {}, acc3 = {};
    int s0 = kc * (kS1 / 16), s1 = s0 + (kS1 / 16);
    for (int s = s0; s < s1; ++s) {
        v2f a = Af[(size_t)s * 32 + lane];
        size_t xb = ((size_t)s * 32 + btg * 4) * 32 + lane;
        v2f b0 = Xf[xb];
        v2f b1 = Xf[xb + 32];
        v2f b2 = Xf[xb + 64];
        v2f b3 = Xf[xb + 96];
        acc0 = __builtin_amdgcn_wmma_f32_16x16x4_f32(false, a, false, b0, (short)0, acc0, false, false);
        acc1 = __builtin_amdgcn_wmma_f32_16x16x4_f32(false, a, false, b1, (short)0, acc1, false, false);
        acc2 = __builtin_amdgcn_wmma_f32_16x16x4_f32(false, a, false, b2, (short)0, acc2, false, false);
        acc3 = __builtin_amdgcn_wmma_f32_16x16x4_f32(false, a, false, b3, (short)0, acc3, false, false);
    }

    // D layout: lane(0-15)=N, VGPR j -> M=j ; lane(16-31) -> M=j+8
    int col  = lane & 15;
    int orow = (lane >> 4) << 3;
    float* Sp = s_t + (size_t)u * kO * kB;
    int bbase = btg * 64;
#pragma unroll
    for (int j = 0; j < 8; ++j) {
        float* row = Sp + (size_t)(orow + j) * kB + bbase + col;
        atomicAdd(row +  0, acc0[j]);
        atomicAdd(row + 16, acc1[j]);
        atomicAdd(row + 32, acc2[j]);
        atomicAdd(row + 48, acc3[j]);
    }
}

// ---------------------------------------------------------------------------
// squash over O: v = (n/(1+n)) * s / sqrt(n).
// Writes v directly in GEMM2 A-fragment order:
//   vfrag[u][bs][lane][j] = v[b,u,o],  lane = o + 16*((b>>1)&1), j = b&1, bs = b>>2
// On the last iteration also writes d_out[b,u,o].
// ---------------------------------------------------------------------------
__global__ void squash_kernel(const float* __restrict__ s_t, float* __restrict__ vfrag,
                              float* __restrict__ out, int write_out) {
    int t = blockIdx.x * blockDim.x + threadIdx.x;   // t = u*kB + b
    if (t >= kU * kB) return;
    int u = t >> 9;
    int b = t & 511;
    const float* Sp = s_t + (size_t)u * kO * kB;
    float sv[kO];
    float n = 0.0f;
#pragma unroll
    for (int o = 0; o < kO; ++o) { sv[o] = Sp[o * kB + b]; n += sv[o] * sv[o]; }
    float scale = (n / (1.0f + n)) * rsqrtf(n);
    int bs = b >> 2, j = b & 1, hi = (b >> 1) & 1;
    float* Vf = vfrag + (((size_t)u * kS2 + bs) * 32) * 2;
#pragma unroll
    for (int o = 0; o < kO; ++o) {
        float v = sv[o] * scale;
        Vf[(o + 16 * hi) * 2 + j] = v;
        if (write_out) out[(size_t)b * kUO + u * kO + o] = v;
    }
}

// ---------------------------------------------------------------------------
// GEMM2 + fused agree epilogue:
//   G[u](o,k) = sum_b v[b,u,o] * x[b,k]        (M=16 o, N=64 k-slab, K=512 b)
//   P[u][k]   = sum_o G[u](o,k) * W[c][u][o][i]
// A-fragments come packed from squash (1 b64/step); B read straight from x
// (row-contiguous).  shfl_xor(16) folds the two o-halves per column.
// ---------------------------------------------------------------------------
__global__ __launch_bounds__(32) void gemm2_kernel(const float* __restrict__ vfrag,
                                                   const float* __restrict__ x,
                                                   const float* __restrict__ W,
                                                   float* __restrict__ P) {
    int u    = blockIdx.x / (kK / 64);    // grid = 10 * 144 = 1440 waves
    int ktg  = blockIdx.x % (kK / 64);
    int k0   = ktg * 64;
    int lane = threadIdx.x;
    int col  = lane & 15;
    int khalf = (lane >> 4) << 1;

    const v2f* Vf = (const v2f*)vfrag + (size_t)u * kS2 * 32;
    v8f acc[4] = {};
    for (int bs = 0; bs < kS2; ++bs) {
        v2f a = Vf[(size_t)bs * 32 + lane];
        int br = bs * 4 + khalf;
        const float* xr0 = x + (size_t)(br + 0) * kK + k0 + col;
        const float* xr1 = x + (size_t)(br + 1) * kK + k0 + col;
        v2f b0 = {xr0[0],  xr1[0]};
        v2f b1 = {xr0[16], xr1[16]};
        v2f b2 = {xr0[32], xr1[32]};
        v2f b3 = {xr0[48], xr1[48]};
        acc[0] = __builtin_amdgcn_wmma_f32_16x16x4_f32(false, a, false, b0, (short)0, acc[0], false, false);
        acc[1] = __builtin_amdgcn_wmma_f32_16x16x4_f32(false, a, false, b1, (short)0, acc[1], false, false);
        acc[2] = __builtin_amdgcn_wmma_f32_16x16x4_f32(false, a, false, b2, (short)0, acc[2], false, false);
        acc[3] = __builtin_amdgcn_wmma_f32_16x16x4_f32(false, a, false, b3, (short)0, acc[3], false, false);
    }

    int obase = (lane >> 4) << 3;
#pragma unroll
    for (int tile = 0; tile < 4; ++tile) {
        int k = k0 + tile * 16 + col;
        int c = k % kC;
        int i = k / kC;               // constant across each 16-tile (kC % 16 == 0)
        const float* Wp = W + ((size_t)(c * kU + u) * kO) * kI + i;
        float part = 0.0f;
#pragma unroll
        for (int j = 0; j < 8; ++j)
            part += acc[tile][j] * Wp[(size_t)(obase + j) * kI];
        part += __shfl_xor(part, 16, 32);
        if (lane < 16) P[(size_t)u * kK + k] = part;
    }
}

// ---------------------------------------------------------------------------
// b_log[c][u] += (1/B) * sum_i P[u][i*kC + c]
// ---------------------------------------------------------------------------
__global__ void update_b_kernel(const float* __restrict__ P, float* __restrict__ b_log) {
    int t = blockIdx.x * blockDim.x + threadIdx.x;   // t = c*kU + u
    if (t >= kC * kU) return;
    int c = t / kU;
    int u = t - c * kU;
    float agree = 0.0f;
#pragma unroll
    for (int i = 0; i < kI; ++i) agree += P[(size_t)u * kK + i * kC + c];
    b_log[t] += agree * (1.0f / (float)kB);
}

// ---------------------------------------------------------------------------
// Host-side orchestration (all on `stream`, graph-capture safe)
// ---------------------------------------------------------------------------
extern "C" void kernel_launch(void* const* d_in, const int* in_sizes, int n_in,
                              void* d_out, int out_size, void* d_ws, size_t ws_size,
                              hipStream_t stream) {
    const float* x = (const float*)d_in[0];   // [512, 8, 1152] == [B][K] flat
    const float* W = (const float*)d_in[1];   // [1152, 10, 16, 8]
    float* out = (float*)d_out;               // [512, 10, 16, 1]

    // Workspace layout (floats); all offsets 8-byte aligned for v2f loads
    float* ws     = (float*)d_ws;
    size_t off = 0;
    float* xfrag  = ws + off;  off += (size_t)kS1 * 32 * 32 * 2;   // 4,718,592
    float* b_log  = ws + off;  off += kC * kU;                     // 11,520
    float* c_coef = ws + off;  off += kC * kU;                     // 11,520
    float* afrag  = ws + off;  off += (size_t)kU * kS1 * 32 * 2;   // 1,474,560
    float* s_t    = ws + off;  off += kU * kO * kB;                // 81,920
    float* vfrag  = ws + off;  off += (size_t)kU * kS2 * 32 * 2;   // 81,920
    float* P      = ws + off;  off += (size_t)kU * kK;             // 92,160
    // total ~25.9 MB (fits in L2 alongside x and W)

    // One-time: fragment-pack x, zero routing logits
    pack_xfrag_kernel<<<(kS1 * 32 * 32 * 2 + 255) / 256, 256, 0, stream>>>(x, xfrag);
    zero_kernel<<<(kC * kU + 255) / 256, 256, 0, stream>>>(b_log, kC * kU);

    for (int it = 0; it < 3; ++it) {
        softmax_kernel<<<(kC + 63) / 64, 64, 0, stream>>>(b_log, c_coef);
        pack_afrag_kernel<<<(kU * kS1 * 64 + 255) / 256, 256, 0, stream>>>(W, c_coef, afrag);
        zero_kernel<<<(kU * kO * kB + 255) / 256, 256, 0, stream>>>(s_t, kU * kO * kB);
        gemm1_kernel<<<kU * 8 * 16, 32, 0, stream>>>(afrag, xfrag, s_t);
        squash_kernel<<<(kU * kB + 63) / 64, 64, 0, stream>>>(s_t, vfrag, out, it == 2 ? 1 : 0);
        if (it < 2) {   // last iteration's agree/b-update is dead in the reference
            gemm2_kernel<<<kU * (kK / 64), 32, 0, stream>>>(vfrag, x, W, P);
            update_b_kernel<<<(kC * kU + 63) / 64, 64, 0, stream>>>(P, b_log);
        }
    }
}